// ROIGenerator_11476152615314
// MI455X (gfx1250) — compile-verified
//
#include <hip/hip_runtime.h>
#include <stdint.h>
#include <float.h>

#ifndef __has_builtin
#define __has_builtin(x) 0
#endif

// ---- CDNA5 async global->LDS path (probe-verified on this toolchain) ----
#if __has_builtin(__builtin_amdgcn_global_load_async_to_lds_b32)
#define USE_ASYNC_LDS 1
#else
#define USE_ASYNC_LDS 0
#endif

#define NBATCH 8
#define NLEV   4
#define NBL    (NBATCH * NLEV)   // 32 (batch,level) problems
#define KREAL  1000              // pre/post NMS top-k
#define KPAD   1024
#define NTH    1024
#define NWAVE  (NTH / 32)        // 32 waves per block (wave32)
#define FINAL_N 4096             // 4 levels * 1024 padded

// monotonic float->uint key (descending float == descending key)
__device__ __forceinline__ uint32_t f2key(float f) {
  uint32_t b = __float_as_uint(f);
  return (b & 0x80000000u) ? ~b : (b | 0x80000000u);
}
__device__ __forceinline__ float key2f(uint32_t u) {
  uint32_t b = (u & 0x80000000u) ? (u & 0x7fffffffu) : ~u;
  return __uint_as_float(b);
}

__device__ __forceinline__ uint32_t ballot32(bool p) {
#if __has_builtin(__builtin_amdgcn_ballot_w32)
  return __builtin_amdgcn_ballot_w32(p);
#else
  return (uint32_t)__ballot(p);
#endif
}

#if USE_ASYNC_LDS
__device__ __forceinline__ void async_b32(void* lds, const void* g) {
  __builtin_amdgcn_global_load_async_to_lds_b32(
      (__attribute__((address_space(1))) int*)(uintptr_t)g,
      (__attribute__((address_space(3))) int*)lds,
      0, 0);
}
__device__ __forceinline__ void wait_async0() {
#if __has_builtin(__builtin_amdgcn_s_wait_asynccnt)
  __builtin_amdgcn_s_wait_asynccnt(0);
#else
  asm volatile("s_wait_asynccnt 0" ::: "memory");
#endif
}
#endif

// ---------------------------------------------------------------------------
// Kernel 1: per (batch,level) radix-select top-1000 + LDS bitonic sort.
// Wave-privatized histograms (32x256) cut ds-atomic contention ~32x.
// ---------------------------------------------------------------------------
__global__ __launch_bounds__(NTH) void k_topk_select(
    const float* __restrict__ s2, const float* __restrict__ s3,
    const float* __restrict__ s4, const float* __restrict__ s5,
    const float* __restrict__ b2, const float* __restrict__ b3,
    const float* __restrict__ b4, const float* __restrict__ b5,
    float* __restrict__ cand_scores, float* __restrict__ cand_boxes)
{
  const int bl    = blockIdx.x;
  const int batch = bl >> 2;
  const int lvl   = bl & 3;
  const int n = (lvl == 0) ? 196608 : (lvl == 1) ? 49152 : (lvl == 2) ? 12288 : 3072;
  const float* sp = (lvl == 0) ? s2 : (lvl == 1) ? s3 : (lvl == 2) ? s4 : s5;
  const float* bp = (lvl == 0) ? b2 : (lvl == 1) ? b3 : (lvl == 2) ? b4 : b5;
  sp += (size_t)batch * n;
  bp += (size_t)batch * n * 4;

  __shared__ int      hist2[NWAVE][256];   // wave-private histograms (32 KB)
  __shared__ int      hist[256];
  __shared__ uint32_t sh_prefix;
  __shared__ int      sh_kk;
  __shared__ int      sh_cnt, sh_cntEq;
  __shared__ uint32_t skey[KPAD];
  __shared__ int      sidx[KPAD];

  const int tid = threadIdx.x;
  const int wv  = tid >> 5;

  // ---- radix select (MSB -> LSB, 8 bits per pass) for the K-th largest key
  uint32_t prefix = 0;
  int kk = KREAL;
  for (int p = 3; p >= 0; --p) {
    int* h2 = &hist2[0][0];
    for (int d = tid; d < NWAVE * 256; d += NTH) h2[d] = 0;
    __syncthreads();
    const int shift = p * 8;
    for (int e = tid; e < n; e += NTH) {
      uint32_t u = f2key(sp[e]);
      bool match = (p == 3) || (((u ^ prefix) >> (shift + 8)) == 0);
      if (match) atomicAdd(&hist2[wv][(u >> shift) & 255], 1);
    }
    __syncthreads();
    if (tid < 256) {
      int s = 0;
      for (int w = 0; w < NWAVE; ++w) s += hist2[w][tid];
      hist[tid] = s;
    }
    __syncthreads();
    if (tid == 0) {
      int cum = 0, sel = 0;
      for (int d = 255; d >= 0; --d) {
        int h = hist[d];
        if (cum + h >= kk) { sel = d; break; }
        cum += h;
      }
      sh_prefix = prefix | ((uint32_t)sel << shift);
      sh_kk = kk - cum;
    }
    __syncthreads();
    prefix = sh_prefix;
    kk = sh_kk;
    __syncthreads();
  }
  const uint32_t T = prefix;   // K-th largest key
  const int needEq = kk;       // # of elements == T we must take

  // ---- compact candidates: keys > T, then keys == T (ties ~impossible)
  if (tid == 0) { sh_cnt = 0; sh_cntEq = 0; }
  __syncthreads();
  for (int e = tid; e < n; e += NTH) {
    uint32_t u = f2key(sp[e]);
    if (u > T) {
      int q = atomicAdd(&sh_cnt, 1);      // ds_add_rtn_u32
      if (q < KREAL) { skey[q] = u; sidx[q] = e; }
    }
  }
  __syncthreads();
  const int gBase = sh_cnt;               // == KREAL - needEq
  for (int e = tid; e < n; e += NTH) {
    uint32_t u = f2key(sp[e]);
    if (u == T) {
      int q = atomicAdd(&sh_cntEq, 1);
      if (q < needEq) { skey[gBase + q] = u; sidx[gBase + q] = e; }
    }
  }
  __syncthreads();
  if (tid >= KREAL) { skey[tid] = 0u; sidx[tid] = 0x7fffffff; }
  __syncthreads();

  // ---- bitonic sort 1024 elems, (key desc, idx asc) => deterministic
  for (unsigned k = 2; k <= KPAD; k <<= 1) {
    for (unsigned j = k >> 1; j > 0; j >>= 1) {
      unsigned i = (unsigned)tid;
      unsigned ixj = i ^ j;
      if (ixj > i) {
        uint32_t ka = skey[i], kb = skey[ixj];
        int ia = sidx[i], ib = sidx[ixj];
        bool before = (ka > kb) || (ka == kb && ia < ib);
        bool dir = ((i & k) == 0);
        if (before != dir) {
          skey[i] = kb; skey[ixj] = ka;
          sidx[i] = ib; sidx[ixj] = ia;
        }
      }
      __syncthreads();
    }
  }

  // ---- emit sorted scores + gathered boxes (pad slots: -FLT_MAX / zeros)
  {
    float* cs = cand_scores + (size_t)bl * KPAD;
    float* cb = cand_boxes + (size_t)bl * KPAD * 4;
    if (tid < KREAL) {
      int id = sidx[tid];
      cs[tid] = sp[id];
      float4 bx = *(const float4*)(bp + (size_t)id * 4);
      ((float4*)cb)[tid] = bx;
    } else {
      cs[tid] = -FLT_MAX;
      ((float4*)cb)[tid] = make_float4(0.f, 0.f, 0.f, 0.f);
    }
  }
}

// ---------------------------------------------------------------------------
// Kernel 2: bitmask NMS. Phase A (fully parallel, 1 barrier): build the
// 1000x1024-bit suppression matrix in LDS — thread t owns column box t in
// registers; a wave32 ballot over 32 columns IS one matrix word. Phase B
// (single-wave serial): wave 0 holds the 1024-bit removed vector as one
// uint32/lane, resolves row i via one __shfl broadcast + conditional OR
// (next row prefetched to hide ds_load latency). Then stable compaction.
// Uses the CDNA5 320KB WGP LDS: matrix 128000 B + boxes SoA 24 KB.
// ---------------------------------------------------------------------------
__global__ __launch_bounds__(NTH) void k_nms(
    const float* __restrict__ cand_scores, const float* __restrict__ cand_boxes,
    float* __restrict__ lvl_scores, float* __restrict__ lvl_boxes)
{
  const int bl = blockIdx.x;
  const int tid = threadIdx.x;
  const int wv = tid >> 5;
  const int lane = tid & 31;

  __shared__ float    sy1[KPAD], sx1[KPAD], sy2[KPAD], sx2[KPAD], sar[KPAD], ssc[KPAD];
  __shared__ uint32_t supbits[KREAL * 32];   // row i, word w -> columns [32w,32w+31]
  __shared__ uint32_t sval[NWAVE];           // validity bits (score>0, in range)
  __shared__ uint32_t skeep[NWAVE];          // final keep bits
  __shared__ int      scn[KPAD];             // inclusive scan for compaction

  const float* cb = cand_boxes + ((size_t)bl * KPAD + tid) * 4;
  const float* cs = cand_scores + (size_t)bl * KPAD + tid;
#if USE_ASYNC_LDS
  async_b32(&sy1[tid], cb + 0);
  async_b32(&sx1[tid], cb + 1);
  async_b32(&sy2[tid], cb + 2);
  async_b32(&sx2[tid], cb + 3);
  async_b32(&ssc[tid], cs);
  wait_async0();               // this wave's async LDS writes complete
#else
  sy1[tid] = cb[0]; sx1[tid] = cb[1]; sy2[tid] = cb[2]; sx2[tid] = cb[3];
  ssc[tid] = *cs;
#endif
  __syncthreads();

  // column box in registers (thread t <-> column j = t)
  const float jy1 = sy1[tid], jx1 = sx1[tid], jy2 = sy2[tid], jx2 = sx2[tid];
  const float jar = (jy2 - jy1) * (jx2 - jx1);
  sar[tid] = jar;
  const bool jvalid = (tid < KREAL) && (ssc[tid] > 0.0f);
  uint32_t vb = ballot32(jvalid);
  if (lane == 0) sval[wv] = vb;
  __syncthreads();

  // ---- Phase A: suppression matrix, fully parallel (1000 rows)
  for (int i = 0; i < KREAL; ++i) {
    float iy1b = sy1[i], ix1b = sx1[i], iy2b = sy2[i], ix2b = sx2[i]; // broadcast
    float iarb = sar[i];
    float cy1 = fmaxf(iy1b, jy1);
    float cx1 = fmaxf(ix1b, jx1);
    float cy2 = fminf(iy2b, jy2);
    float cx2 = fminf(ix2b, jx2);
    float ih = fmaxf(cy2 - cy1, 0.0f);
    float iw = fmaxf(cx2 - cx1, 0.0f);
    float inter = ih * iw;
    float uni = iarb + jar - inter;
    float iou = (uni > 0.0f) ? (inter / uni) : 0.0f;
    uint32_t rowword = ballot32(iou > 0.7f);
    if (lane == 0) supbits[i * 32 + wv] = rowword;
  }
  __syncthreads();

  // ---- Phase B: serial resolution by wave 0 only (register-resident)
  if (tid < 32) {
    uint32_t remv = ~sval[lane];             // lane l: removed bits for j in [32l,32l+32)
    uint32_t kpw = 0;
    uint32_t row = supbits[0 * 32 + lane];
    for (int i = 0; i < KREAL; ++i) {
      uint32_t nrow = (i + 1 < KREAL) ? supbits[(i + 1) * 32 + lane] : 0u; // prefetch
      uint32_t w = __shfl(remv, i >> 5, 32); // broadcast word holding bit i
      bool kept = ((w >> (i & 31)) & 1u) == 0u;   // uniform across wave
      if (lane == (i >> 5)) kpw |= (uint32_t)kept << (i & 31);
      if (kept) remv |= row;                 // suppress row i's victims
      row = nrow;
    }
    skeep[lane] = kpw;
  }
  __syncthreads();

  // ---- stable compaction: kept boxes first (score order), zeros after
  int f = (tid < KREAL) ? (int)((skeep[tid >> 5] >> (tid & 31)) & 1u) : 0;
  scn[tid] = f;
  __syncthreads();
  for (int off = 1; off < KPAD; off <<= 1) {
    int add = (tid >= off) ? scn[tid - off] : 0;
    __syncthreads();
    scn[tid] += add;
    __syncthreads();
  }

  float* os = lvl_scores + (size_t)bl * KPAD;
  float* ob = lvl_boxes + (size_t)bl * KPAD * 4;
  os[tid] = (tid < KREAL) ? 0.0f : -FLT_MAX;        // masked slots are exact 0.0
  ((float4*)ob)[tid] = make_float4(0.f, 0.f, 0.f, 0.f);
  __syncthreads();
  if (f) {
    int pz = scn[tid] - 1;
    os[pz] = ssc[tid];
    ((float4*)ob)[pz] = make_float4(jy1, jx1, jy2, jx2);
  }
}

// ---------------------------------------------------------------------------
// Kernel 3: per-batch top-1000 over 4x1000 candidates via 4096-elem LDS
// bitonic sort (4 elems/thread). Index tie-break => deterministic; all ties
// are masked 0.0 slots whose rois are zero, matching the reference exactly.
// ---------------------------------------------------------------------------
__global__ __launch_bounds__(NTH) void k_final(
    const float* __restrict__ lvl_scores, const float* __restrict__ lvl_boxes,
    float* __restrict__ out_rois, float* __restrict__ out_scores)
{
  const int batch = blockIdx.x;
  const int tid = threadIdx.x;
  __shared__ uint32_t fkey[FINAL_N];
  __shared__ int      fpay[FINAL_N];

  for (int e = tid; e < FINAL_N; e += NTH) {
    int lv = e >> 10, r = e & 1023;
    int slot = (batch * NLEV + lv) * KPAD + r;
    uint32_t key = 0u;                       // pad slots sort to the bottom
    if (r < KREAL) key = f2key(lvl_scores[slot]);
    fkey[e] = key;
    fpay[e] = slot;
  }
  __syncthreads();

  for (unsigned k = 2; k <= FINAL_N; k <<= 1) {
    for (unsigned j = k >> 1; j > 0; j >>= 1) {
#pragma unroll
      for (int s = 0; s < FINAL_N / NTH; ++s) {
        unsigned i = (unsigned)tid + (unsigned)s * NTH;
        unsigned ixj = i ^ j;
        if (ixj > i) {
          uint32_t ka = fkey[i], kb = fkey[ixj];
          int ia = fpay[i], ib = fpay[ixj];
          bool before = (ka > kb) || (ka == kb && ia < ib);
          bool dir = ((i & k) == 0);
          if (before != dir) {
            fkey[i] = kb; fkey[ixj] = ka;
            fpay[i] = ib; fpay[ixj] = ia;
          }
        }
      }
      __syncthreads();
    }
  }

  if (tid < KREAL) {
    uint32_t key = fkey[tid];
    int pay = fpay[tid];
    out_scores[(size_t)batch * KREAL + tid] = key2f(key);
    float4 b = ((const float4*)lvl_boxes)[pay];
    ((float4*)out_rois)[(size_t)batch * KREAL + tid] = b;
  }
}

// ---------------------------------------------------------------------------
extern "C" void kernel_launch(void* const* d_in, const int* in_sizes, int n_in,
                              void* d_out, int out_size, void* d_ws, size_t ws_size,
                              hipStream_t stream)
{
  (void)in_sizes; (void)n_in; (void)out_size; (void)ws_size;

  // setup_inputs() order: boxes_p2, scores_p2, boxes_p3, scores_p3, ...
  const float* b2 = (const float*)d_in[0];
  const float* s2 = (const float*)d_in[1];
  const float* b3 = (const float*)d_in[2];
  const float* s3 = (const float*)d_in[3];
  const float* b4 = (const float*)d_in[4];
  const float* s4 = (const float*)d_in[5];
  const float* b5 = (const float*)d_in[6];
  const float* s5 = (const float*)d_in[7];

  // workspace: 32*1024 scores + 32*1024*4 boxes = 640 KB (reused in place by k_nms)
  float* cand_scores = (float*)d_ws;
  float* cand_boxes  = cand_scores + (size_t)NBL * KPAD;

  float* out_rois   = (float*)d_out;                         // (8,1000,4)
  float* out_scores = out_rois + (size_t)NBATCH * KREAL * 4; // (8,1000)

  k_topk_select<<<NBL, NTH, 0, stream>>>(s2, s3, s4, s5, b2, b3, b4, b5,
                                         cand_scores, cand_boxes);
  k_nms<<<NBL, NTH, 0, stream>>>(cand_scores, cand_boxes,
                                 cand_scores, cand_boxes);   // in-place
  k_final<<<NBATCH, NTH, 0, stream>>>(cand_scores, cand_boxes,
                                      out_rois, out_scores);
}